// HybridBlock_15908558865316
// MI455X (gfx1250) — compile-verified
//
#include <hip/hip_runtime.h>
#include <hip/hip_bf16.h>

typedef float v2f __attribute__((ext_vector_type(2)));
typedef float v4f __attribute__((ext_vector_type(4)));
typedef float v8f __attribute__((ext_vector_type(8)));

#define BM 128
#define BN 32
#define BK 16
#define APAD 4   // A row stride = BK+APAD = 20 floats -> 16B-aligned rows, conflict-free column reads
#define BPAD 4   // B row stride = BN+BPAD = 36 floats -> 16B-aligned rows

// ---------------------------------------------------------------------------
// LayerNorm: one 256-thread block per row of (M, Dm)
// ---------------------------------------------------------------------------
__global__ __launch_bounds__(256) void ln_kernel(const float* __restrict__ x,
                                                 const float* __restrict__ g,
                                                 const float* __restrict__ b,
                                                 float* __restrict__ out, int Dm) {
  const int row = blockIdx.x;
  const float* xr = x + (size_t)row * Dm;
  float s = 0.f, s2 = 0.f;
  for (int i = threadIdx.x; i < Dm; i += 256) {
    float v = xr[i];
    s += v;
    s2 += v * v;
  }
  for (int off = 16; off > 0; off >>= 1) {
    s  += __shfl_xor(s,  off, 32);
    s2 += __shfl_xor(s2, off, 32);
  }
  __shared__ float red0[8], red1[8];
  const int wave = threadIdx.x >> 5, lane = threadIdx.x & 31;
  if (lane == 0) { red0[wave] = s; red1[wave] = s2; }
  __syncthreads();
  s = 0.f; s2 = 0.f;
#pragma unroll
  for (int w = 0; w < 8; ++w) { s += red0[w]; s2 += red1[w]; }
  const float mean = s / (float)Dm;
  const float var  = s2 / (float)Dm - mean * mean;
  const float inv  = rsqrtf(var + 1e-5f);
  float* orow = out + (size_t)row * Dm;
  for (int i = threadIdx.x; i < Dm; i += 256) {
    orow[i] = (xr[i] - mean) * inv * g[i] + b[i];
  }
}

// ---------------------------------------------------------------------------
// FP32 WMMA GEMM: C = A(MxK) * B(KxN). Block tile BM=128 x BN=32, BK=16.
// Requires: M % 128 == 0, N % 32 == 0, K % 16 == 0 (true for all call sites;
// no bounds checks -> straight-line b128 staging).
// 128 threads = 4 waves; wave w owns rows [w*32, w*32+32): 2 M-subtiles x
// 2 N-halves = 4 accumulators -> 16 v_wmma per K-step.
// mode 0: C = AB
// mode 1: C = softplus(AB + bias[col])
// mode 2: C = AB + res
// ---------------------------------------------------------------------------
__global__ __launch_bounds__(128) void gemm_wmma_f32(
    const float* __restrict__ A, int lda,
    const float* __restrict__ B, int ldb,
    float* __restrict__ C, int ldc,
    int M, int N, int K, int mode,
    const float* __restrict__ bias, const float* __restrict__ res) {
  __shared__ float As[BM][BK + APAD];   // 128 x 20 floats = 10 KB
  __shared__ float Bs[BK][BN + BPAD];   //  16 x 36 floats = 2.25 KB

  const int tid  = threadIdx.x;
  const int wave = tid >> 5;
  const int lane = tid & 31;
  const size_t m_blk = (size_t)blockIdx.y * BM;
  const size_t n_blk = (size_t)blockIdx.x * BN;

  const int ln   = lane & 15;            // A: M index | B/C/D: N index
  const int kh   = (lane >> 4) << 1;     // lane half -> K pair {0,1} / {2,3}
  const int wrow = wave << 5;            // wave's 32-row M offset in tile

  // staging coordinates (b128 = 4 floats per transaction)
  const int ar = tid >> 2;               // A: 4 float4 per 16-wide row
  const int ac = (tid & 3) << 2;
  const int br = tid >> 3;               // B: 8 float4 per 32-wide row
  const int bc = (tid & 7) << 2;

  v8f acc00 = {}, acc01 = {}, acc10 = {}, acc11 = {};

  for (int k0 = 0; k0 < K; k0 += BK) {
    // ---- stage A tile 128x16: 512 float4, 4 per thread ----
#pragma unroll
    for (int i = 0; i < 4; ++i) {
      int r = ar + i * 32;               // 32 rows covered per pass
      const v4f v = *(const v4f*)(A + (m_blk + r) * lda + k0 + ac);
      *(v4f*)&As[r][ac] = v;
    }
    // ---- stage B tile 16x32: 128 float4, 1 per thread ----
    {
      const v4f v = *(const v4f*)(B + (size_t)(k0 + br) * ldb + n_blk + bc);
      *(v4f*)&Bs[br][bc] = v;
    }
    // prefetch next K-tile while this one computes
    if (k0 + BK < K) {
      __builtin_prefetch(A + (m_blk + ar) * lda + k0 + BK + ac, 0, 1);
      __builtin_prefetch(B + (size_t)(k0 + BK + br) * ldb + n_blk + bc, 0, 1);
    }
    __syncthreads();

#pragma unroll
    for (int kk = 0; kk < BK; kk += 4) {
      v2f a0, a1, b0, b1;
      a0[0] = As[wrow + ln][kk + kh];
      a0[1] = As[wrow + ln][kk + kh + 1];
      a1[0] = As[wrow + 16 + ln][kk + kh];
      a1[1] = As[wrow + 16 + ln][kk + kh + 1];
      b0[0] = Bs[kk + kh][ln];
      b0[1] = Bs[kk + kh + 1][ln];
      b1[0] = Bs[kk + kh][16 + ln];
      b1[1] = Bs[kk + kh + 1][16 + ln];
      acc00 = __builtin_amdgcn_wmma_f32_16x16x4_f32(false, a0, false, b0,
                                                    (short)0, acc00, false, false);
      acc01 = __builtin_amdgcn_wmma_f32_16x16x4_f32(false, a0, false, b1,
                                                    (short)0, acc01, false, false);
      acc10 = __builtin_amdgcn_wmma_f32_16x16x4_f32(false, a1, false, b0,
                                                    (short)0, acc10, false, false);
      acc11 = __builtin_amdgcn_wmma_f32_16x16x4_f32(false, a1, false, b1,
                                                    (short)0, acc11, false, false);
    }
    __syncthreads();
  }

  // C/D layout: VGPR i -> M = i + 8*(lane>=16); lane&15 -> N
  const int mhalf = (lane >> 4) << 3;
#pragma unroll
  for (int sub = 0; sub < 2; ++sub) {
#pragma unroll
    for (int half = 0; half < 2; ++half) {
      const v8f& acc = sub == 0 ? (half == 0 ? acc00 : acc01)
                                : (half == 0 ? acc10 : acc11);
      const size_t gn = n_blk + half * 16 + ln;
#pragma unroll
      for (int i = 0; i < 8; ++i) {
        const size_t gm = m_blk + wrow + sub * 16 + mhalf + i;
        float v = acc[i];
        if (mode == 1) {
          v += bias[gn];
          v = (v > 20.f) ? v : log1pf(__expf(v));     // softplus
        } else if (mode == 2) {
          v += res[gm * ldc + gn];
        }
        C[gm * ldc + gn] = v;
      }
    }
  }
}

// ---------------------------------------------------------------------------
// Depthwise causal conv (k=4) + SiLU.  u_raw lives in xz[..., 0:Din].
// ---------------------------------------------------------------------------
__global__ __launch_bounds__(256) void conv_silu_kernel(
    const float* __restrict__ xz, const float* __restrict__ w,
    const float* __restrict__ bias, float* __restrict__ u,
    int L, int Din, int ldxz, long total) {
  long i = (long)blockIdx.x * 256 + threadIdx.x;
  if (i >= total) return;
  int d = (int)(i % Din);
  long row = i / Din;
  int l = (int)(row % L);
  float acc = bias[d];
#pragma unroll
  for (int j = 0; j < 4; ++j) {
    int ll = l - 3 + j;
    if (ll >= 0) acc += xz[(size_t)(row - 3 + j) * ldxz + d] * w[d * 4 + j];
  }
  float sig = 1.f / (1.f + __expf(-acc));
  u[(size_t)row * Din + d] = acc * sig;                 // SiLU
}

// ---------------------------------------------------------------------------
// Selective scan fused with +u*D and SiLU(z) gating.
// One thread per (b,d): 16 states in registers, sequential over L.
// B/C staged through LDS in chunks of 8 timesteps.
// ---------------------------------------------------------------------------
__global__ __launch_bounds__(256) void scan_kernel(
    const float* __restrict__ u, const float* __restrict__ delta,
    const float* __restrict__ dbc, const float* __restrict__ A_log,
    const float* __restrict__ Dv, const float* __restrict__ xz,
    float* __restrict__ yg, int L, int Din) {
  const int blocksPerBatch = Din / 256;
  const int b = blockIdx.x / blocksPerBatch;
  const int d = (blockIdx.x % blocksPerBatch) * 256 + threadIdx.x;

  float a[16], h[16];
#pragma unroll
  for (int n = 0; n < 16; ++n) {
    a[n] = -__expf(A_log[d * 16 + n]);   // A = -exp(A_log)
    h[n] = 0.f;
  }
  const float Dd = Dv[d];

  __shared__ float bcs[8][32];           // [timestep in chunk][B(16) | C(16)]
  const size_t rowbase = (size_t)b * L;
  const int ldxz = 2 * Din;

  for (int l0 = 0; l0 < L; l0 += 8) {
    {
      int t = threadIdx.x >> 5;
      int c = threadIdx.x & 31;
      bcs[t][c] = dbc[(rowbase + l0 + t) * 96 + 64 + c];
    }
    __syncthreads();
#pragma unroll
    for (int t = 0; t < 8; ++t) {
      const size_t row = rowbase + l0 + t;
      const float dt  = delta[row * Din + d];
      const float uu  = u[row * Din + d];
      const float dtu = dt * uu;
      float y = 0.f;
#pragma unroll
      for (int n = 0; n < 16; ++n) {
        float dA = __expf(dt * a[n]);
        h[n] = dA * h[n] + dtu * bcs[t][n];
        y += h[n] * bcs[t][16 + n];
      }
      const float z = xz[row * (size_t)ldxz + Din + d];
      const float sig = 1.f / (1.f + __expf(-z));
      yg[row * Din + d] = (y + uu * Dd) * (z * sig);
    }
    __syncthreads();
  }
}

// ---------------------------------------------------------------------------
extern "C" void kernel_launch(void* const* d_in, const int* in_sizes, int n_in,
                              void* d_out, int out_size, void* d_ws, size_t ws_size,
                              hipStream_t stream) {
  const float* x         = (const float*)d_in[0];
  const float* in_proj_w = (const float*)d_in[1];
  const float* conv_w    = (const float*)d_in[2];
  const float* conv_b    = (const float*)d_in[3];
  const float* x_proj_w  = (const float*)d_in[4];
  const float* dt_proj_w = (const float*)d_in[5];
  const float* dt_proj_b = (const float*)d_in[6];
  const float* A_log     = (const float*)d_in[7];
  const float* Dv        = (const float*)d_in[8];
  const float* out_proj_w= (const float*)d_in[9];
  const float* ln_g      = (const float*)d_in[10];
  const float* ln_b      = (const float*)d_in[11];
  float* out = (float*)d_out;

  const int B = 2, L = 2048, Dm = 1024;
  const int Din = 2 * Dm;            // 2048
  const int dt_rank = Dm / 16;       // 64
  const int M = B * L;               // 4096

  // workspace carve-up (floats)
  float* ws = (float*)d_ws;
  size_t o = 0;
  float* xn    = ws + o; o += (size_t)M * Dm;        // 4096x1024
  float* xz    = ws + o; o += (size_t)M * 2 * Din;   // 4096x4096 (u_raw | z)
  float* u     = ws + o; o += (size_t)M * Din;       // 4096x2048
  float* dbc   = ws + o; o += (size_t)M * 96;        // 4096x96
  float* delta = ws + o; o += (size_t)M * Din;       // 4096x2048
  float* yg    = ws + o;                             // 4096x2048

  // 1) LayerNorm
  ln_kernel<<<M, 256, 0, stream>>>(x, ln_g, ln_b, xn, Dm);

  // 2) xz = xn @ in_proj_w   (M=4096, K=1024, N=4096)
  {
    dim3 grid((2 * Din) / BN, M / BM);
    gemm_wmma_f32<<<grid, 128, 0, stream>>>(xn, Dm, in_proj_w, 2 * Din,
                                            xz, 2 * Din, M, 2 * Din, Dm,
                                            0, nullptr, nullptr);
  }

  // 3) u = SiLU(causal depthwise conv(u_raw))
  {
    long total = (long)M * Din;
    conv_silu_kernel<<<(int)((total + 255) / 256), 256, 0, stream>>>(
        xz, conv_w, conv_b, u, L, Din, 2 * Din, total);
  }

  // 4) dbc = u @ x_proj_w    (M=4096, K=2048, N=96)
  {
    dim3 grid(96 / BN, M / BM);
    gemm_wmma_f32<<<grid, 128, 0, stream>>>(u, Din, x_proj_w, 96,
                                            dbc, 96, M, 96, Din,
                                            0, nullptr, nullptr);
  }

  // 5) delta = softplus(dt @ dt_proj_w + dt_proj_b)  (M=4096, K=64, N=2048)
  //    dt is dbc[:, 0:64] -> A = dbc with lda = 96
  {
    dim3 grid(Din / BN, M / BM);
    gemm_wmma_f32<<<grid, 128, 0, stream>>>(dbc, 96, dt_proj_w, Din,
                                            delta, Din, M, Din, dt_rank,
                                            1, dt_proj_b, nullptr);
  }

  // 6) selective scan + (y + u*D) * SiLU(z)
  scan_kernel<<<B * (Din / 256), 256, 0, stream>>>(u, delta, dbc, A_log, Dv,
                                                   xz, yg, L, Din);

  // 7) out = x + yg @ out_proj_w  (M=4096, K=2048, N=1024)
  {
    dim3 grid(Dm / BN, M / BM);
    gemm_wmma_f32<<<grid, 128, 0, stream>>>(yg, Din, out_proj_w, Dm,
                                            out, Dm, M, Dm, Din,
                                            2, nullptr, x);
  }
}